// MeanAggregator_80204219286419
// MI455X (gfx1250) — compile-verified
//
#include <hip/hip_runtime.h>
#include <stdint.h>

typedef float v4f __attribute__((ext_vector_type(4)));

#define WAVES_PER_BLOCK 8
#define DFIX 128      // feature dim (floats) for the fast path
#define ROWB 512      // bytes per feature row (DFIX * 4)
#define KFIX 16       // neighbors per node for the fast path

// One wave32 per output row. Lane l owns bytes [16*l, 16*l+16) of the 512B row.
// Neighbor rows are gathered with CDNA5 async global->LDS B128 ops (ASYNCcnt),
// consumed in two pipelined chunks via staged s_wait_asynccnt.
__global__ __launch_bounds__(32 * WAVES_PER_BLOCK)
void sage_mean_k16_d128(const float* __restrict__ features,
                        const int* __restrict__ nodes,
                        const int* __restrict__ neigh,
                        float* __restrict__ out,
                        int B, float rcpK)
{
    __shared__ __align__(16) float stage[WAVES_PER_BLOCK][KFIX][DFIX]; // 64 KB

    const int lane = threadIdx.x & 31;
    const int wid  = threadIdx.x >> 5;
    int b = blockIdx.x * WAVES_PER_BLOCK + wid;
    if (b >= B) return;
    b = __builtin_amdgcn_readfirstlane(b);   // wave-uniform: enables SGPR addressing

    // LDS byte offset of this wave's staging buffer, plus this lane's column.
    // (generic pointer to __shared__ truncates to the LDS byte offset)
    const uint32_t lds0 =
        (uint32_t)(uintptr_t)(&stage[wid][0][0]) + (uint32_t)lane * 16u;

    // 16 wave-uniform neighbor indices (4 x b128 loads)
    const int4* __restrict__ nrow = (const int4*)(neigh + (size_t)b * KFIX);
    int4 i0 = nrow[0], i1 = nrow[1], i2 = nrow[2], i3 = nrow[3];
    int idx[KFIX] = { i0.x, i0.y, i0.z, i0.w,
                      i1.x, i1.y, i1.z, i1.w,
                      i2.x, i2.y, i2.z, i2.w,
                      i3.x, i3.y, i3.z, i3.w };

    // Kick off all 16 row gathers: each is one wave-wide 512B async transfer.
    #pragma unroll
    for (int k = 0; k < KFIX; ++k) {
        uint32_t voff = (uint32_t)idx[k] * (uint32_t)ROWB + (uint32_t)lane * 16u;
        uint32_t ldsa = lds0 + (uint32_t)k * (uint32_t)ROWB;
        asm volatile("global_load_async_to_lds_b128 %0, %1, %2"
                     :: "v"(ldsa), "v"(voff), "s"(features)
                     : "memory");
    }

    // Self row straight to VGPRs, overlapped with the async gathers.
    const int self_row = nodes[b];
    const v4f* __restrict__ frow =
        (const v4f*)(features + (size_t)self_row * DFIX);
    v4f self = frow[lane];

    v4f acc = {0.0f, 0.0f, 0.0f, 0.0f};

    // Async loads complete in order: after asynccnt<=8 the first 8 rows are in LDS.
    asm volatile("s_wait_asynccnt 0x8" ::: "memory");
    #pragma unroll
    for (int k = 0; k < 8; ++k) {
        v4f v = ((const v4f*)(&stage[wid][k][0]))[lane];
        acc += v;
    }
    asm volatile("s_wait_asynccnt 0x0" ::: "memory");
    #pragma unroll
    for (int k = 8; k < KFIX; ++k) {
        v4f v = ((const v4f*)(&stage[wid][k][0]))[lane];
        acc += v;
    }

    v4f r = self + acc * rcpK;

    // Streaming output: non-temporal so the 51MB store doesn't pollute L2
    // (L2 is holding hot feature rows).
    __builtin_nontemporal_store(r, ((v4f*)out) + (size_t)b * (DFIX / 4) + lane);
}

// Generic fallback for shapes other than K=16, D=128 (not the benchmark path).
__global__ void sage_mean_generic(const float* __restrict__ features,
                                  const int* __restrict__ nodes,
                                  const int* __restrict__ neigh,
                                  float* __restrict__ out,
                                  int B, int D, int K, float rcpK)
{
    long long t = (long long)blockIdx.x * blockDim.x + threadIdx.x;
    long long total = (long long)B * D;
    if (t >= total) return;
    int b = (int)(t / D);
    int d = (int)(t % D);
    float acc = 0.0f;
    for (int k = 0; k < K; ++k) {
        int r = neigh[(size_t)b * K + k];
        acc += features[(size_t)r * D + d];
    }
    out[t] = features[(size_t)nodes[b] * D + d] + acc * rcpK;
}

extern "C" void kernel_launch(void* const* d_in, const int* in_sizes, int n_in,
                              void* d_out, int out_size, void* d_ws, size_t ws_size,
                              hipStream_t stream)
{
    const float* features = (const float*)d_in[0];
    const int*   nodes    = (const int*)d_in[1];
    const int*   neigh    = (const int*)d_in[2];
    float*       out      = (float*)d_out;

    const int B = in_sizes[1];                 // batch (number of output rows)
    const int K = in_sizes[2] / B;             // neighbors per node
    const int D = out_size / B;                // feature dim
    const float rcpK = 1.0f / (float)K;

    if (K == KFIX && D == DFIX) {
        const int blocks = (B + WAVES_PER_BLOCK - 1) / WAVES_PER_BLOCK;
        sage_mean_k16_d128<<<blocks, 32 * WAVES_PER_BLOCK, 0, stream>>>(
            features, nodes, neigh, out, B, rcpK);
    } else {
        const long long total = (long long)B * D;
        const int blocks = (int)((total + 255) / 256);
        sage_mean_generic<<<blocks, 256, 0, stream>>>(
            features, nodes, neigh, out, B, D, K, rcpK);
    }
}